// NCutsLoss_644245094904
// MI455X (gfx1250) — compile-verified
//
#include <hip/hip_runtime.h>
#include <stdint.h>

typedef uint32_t u32;
typedef unsigned int __attribute__((ext_vector_type(4))) tdm_u32x4;
typedef int          __attribute__((ext_vector_type(4))) tdm_i32x4;
typedef int          __attribute__((ext_vector_type(8))) tdm_i32x8;
typedef float        __attribute__((ext_vector_type(2))) v2f;
typedef float        __attribute__((ext_vector_type(8))) v8f;

#define B_    2
#define K_    4
#define DD    32
#define HH    32
#define WW    32
#define WIN   7
#define W3    343          // 7^3 taps
#define PDIM  38           // padded spatial extent (32 + 2*3)
#define ROW   38           // floats per staged row
#define CLS   (49 * 38)    // 1862 floats per class in the LDS tile

// ---------------------------------------------------------------------------
// zero the 16 accumulator slots each launch (graph-replay safe)
// ---------------------------------------------------------------------------
__global__ void ncuts_init(float* ws) {
    if (threadIdx.x < 16) ws[threadIdx.x] = 0.0f;
}

// ---------------------------------------------------------------------------
// main streaming-reduction kernel: one block per (b,d,h) row of 32 voxels
// ---------------------------------------------------------------------------
__global__ __launch_bounds__(256)
void ncuts_main(const float* __restrict__ padded_seg,
                const float* __restrict__ weight,
                float* __restrict__ ws) {
    __shared__ __align__(16) float s_seg[K_ * CLS];   // 7448 f32 = 29.8 KB TDM tile
    __shared__ int   s_off[W3];                       // tap -> LDS row offset
    __shared__ float s_red[8][8];                     // [wave][A0..A3,V0..V3]

    const int tid  = threadIdx.x;
    const int lane = tid & 31;
    const int wv   = tid >> 5;            // wave 0..7
    const int h = blockIdx.x, d = blockIdx.y, b = blockIdx.z;

    // ---- issue the TDM load first so the DMA overlaps the table build ------
    if (wv == 0) {
        // TDM 4-D tile: x=38 (w row), y=7 (h+ph), z=7 (d+pd), zz=4 (class)
        unsigned long long gaddr = (unsigned long long)(uintptr_t)padded_seg +
            4ull * ((unsigned long long)b * (K_ * PDIM * PDIM * PDIM) +
                    (unsigned long long)d * (PDIM * PDIM) +
                    (unsigned long long)h * PDIM);
        u32 lds = (u32)(uintptr_t)&s_seg[0];

        tdm_u32x4 g0;
        g0[0] = 1u;                                             // count=1, user mode
        g0[1] = lds;                                            // lds_addr
        g0[2] = (u32)gaddr;                                     // global_addr[31:0]
        g0[3] = ((u32)(gaddr >> 32) & 0x1FFFFFFu) | (2u << 30); // addr[56:32] | type=2

        tdm_i32x8 g1;
        g1[0] = (int)(2u << 16);                 // data_size = 4B; no multicast
        g1[1] = (int)((u32)PDIM << 16);          // tensor_dim0[15:0]=38
        g1[2] = (int)((u32)PDIM << 16);          // dim0[31:16]=0 | tensor_dim1[15:0]=38
        g1[3] = (int)((u32)PDIM << 16);          // dim1[31:16]=0 | tile_dim0=38
        g1[4] = (int)((7u << 16) | 7u);          // tile_dim1=7 | tile_dim2=7
        g1[5] = PDIM;                            // tensor_dim0_stride = 38
        g1[6] = (int)((u32)(PDIM * PDIM) << 16); // stride0 hi=0 | stride1 lo16 = 1444
        g1[7] = 0;                               // stride1 hi = 0

        tdm_i32x4 g2;
        g2[0] = PDIM;                            // tensor_dim2 = 38
        g2[1] = K_;                              // tensor_dim3 = 4
        g2[2] = PDIM * PDIM * PDIM;              // tensor_dim2_stride = 54872 (class)
        g2[3] = (int)((u32)K_ << 16);            // tile_dim3 = 4

        tdm_i32x4 g3 = {0, 0, 0, 0};             // no 5th dim

#if __clang_major__ >= 23
        tdm_i32x8 g4 = {0, 0, 0, 0, 0, 0, 0, 0};
        __builtin_amdgcn_tensor_load_to_lds(g0, g1, g2, g3, g4, 0);
#else
        __builtin_amdgcn_tensor_load_to_lds(g0, g1, g2, g3, 0);
#endif
    }

    // window-offset table: p=(pd,ph,pw) -> (pd*7+ph)*38 + pw  (overlaps the DMA)
    for (int p = tid; p < W3; p += 256) {
        int pd = p / 49;
        int r  = p - pd * 49;
        int ph = r / 7;
        int pw = r - ph * 7;
        s_off[p] = (pd * 7 + ph) * ROW + pw;
    }

    if (wv == 0) __builtin_amdgcn_s_wait_tensorcnt(0);
    __syncthreads();

    // ---- stream the weight tensor (the 90 MB operand), coalesced over taps ----
    float pA[4] = {0, 0, 0, 0};
    float pV[4] = {0, 0, 0, 0};
    const long long wbase0 = (((long long)(b * DD + d) * HH + h) * WW) * W3;

    for (int i = 0; i < 4; ++i) {
        const int v = wv * 4 + i;                     // voxel w-index 0..31
        const float* wp = weight + wbase0 + (long long)v * W3;
        float a0 = 0, a1 = 0, a2 = 0, a3 = 0, av = 0;
        for (int it = 0; it < 11; ++it) {
            int p = it * 32 + lane;
            if (p < W3) {
                float wt  = wp[p];                    // 128B coalesced per wave
                int   off = s_off[p] + v;
                a0 += wt * s_seg[0 * CLS + off];
                a1 += wt * s_seg[1 * CLS + off];
                a2 += wt * s_seg[2 * CLS + off];
                a3 += wt * s_seg[3 * CLS + off];
                av += wt;
            }
        }
        // wave32 butterfly reduction over the 32 taps-lanes
        for (int o = 16; o > 0; o >>= 1) {
            a0 += __shfl_xor(a0, o, 32);
            a1 += __shfl_xor(a1, o, 32);
            a2 += __shfl_xor(a2, o, 32);
            a3 += __shfl_xor(a3, o, 32);
            av += __shfl_xor(av, o, 32);
        }
        // seg[b,k,d,h,v] == padded_seg center (pd=3,ph=3,x=v+3) — read from tile
        const int c = 24 * ROW + v + 3;
        float s0 = s_seg[0 * CLS + c], s1 = s_seg[1 * CLS + c];
        float s2 = s_seg[2 * CLS + c], s3 = s_seg[3 * CLS + c];
        pA[0] += a0 * s0;  pA[1] += a1 * s1;  pA[2] += a2 * s2;  pA[3] += a3 * s3;
        pV[0] += av * s0;  pV[1] += av * s1;  pV[2] += av * s2;  pV[3] += av * s3;
    }

    if (lane == 0) {
        for (int q = 0; q < 4; ++q) { s_red[wv][q] = pA[q]; s_red[wv][4 + q] = pV[q]; }
    }
    __syncthreads();

    if (wv == 0) {
        // Exact f32 WMMA column-sum: A = all-ones (layout-free), B carries the
        // 8x8 partials as B[k=wave&3, n=q+8*(wave>>2)]; D[0,n] = sum_k B[k,n].
        // B 4x16 striping: VGPR0 lanes0-15 -> K=0, lanes16-31 -> K=2;
        //                  VGPR1 lanes0-15 -> K=1, lanes16-31 -> K=3.
        int n  = lane & 15;
        int kk = (lane >> 4) << 1;       // 0 or 2
        int q  = n & 7;
        int wg = (n >> 3) << 2;          // wave group 0 or 4
        v2f bvec;
        bvec.x = s_red[wg + kk + 0][q];
        bvec.y = s_red[wg + kk + 1][q];
        v2f avec; avec.x = 1.0f; avec.y = 1.0f;
        v8f cz = {0, 0, 0, 0, 0, 0, 0, 0};
        v8f dacc = __builtin_amdgcn_wmma_f32_16x16x4_f32(
            false, avec, false, bvec, (short)0, cz, false, false);
        float r0  = dacc[0];                              // row M=0, N=lane (0..15)
        float tot = __shfl(r0, lane & 7, 32) + __shfl(r0, (lane & 7) + 8, 32);
        if (lane < 8) atomicAdd(&ws[b * 8 + lane], tot);
    }
}

// ---------------------------------------------------------------------------
// finalize: out[b] = K - sum_k assocA/assocV
// ---------------------------------------------------------------------------
__global__ void ncuts_finalize(const float* __restrict__ ws, float* __restrict__ out) {
    int b = threadIdx.x;
    if (b < B_) {
        float s = 0.0f;
        for (int k = 0; k < K_; ++k) s += ws[b * 8 + k] / ws[b * 8 + 4 + k];
        out[b] = 4.0f - s;
    }
}

extern "C" void kernel_launch(void* const* d_in, const int* in_sizes, int n_in,
                              void* d_out, int out_size, void* d_ws, size_t ws_size,
                              hipStream_t stream) {
    // inputs: d_in[0]=seg (unused; equals padded_seg center), d_in[1]=padded_seg,
    //         d_in[2]=weight
    const float* padded_seg = (const float*)d_in[1];
    const float* weight     = (const float*)d_in[2];
    float* ws  = (float*)d_ws;
    float* out = (float*)d_out;

    ncuts_init<<<1, 32, 0, stream>>>(ws);
    dim3 grid(HH, DD, B_);                 // h, d, b
    ncuts_main<<<grid, 256, 0, stream>>>(padded_seg, weight, ws);
    ncuts_finalize<<<1, 32, 0, stream>>>(ws, out);
}